// Dictionary_27659589386431
// MI455X (gfx1250) — compile-verified
//
#include <hip/hip_runtime.h>
#include <hip/hip_bf16.h>
#include <math.h>

// CDNA5 / gfx1250, wave32 only.
typedef __attribute__((ext_vector_type(2))) float v2f;   // A/B operand of V_WMMA_F32_16X16X4_F32 (64 f32 / 32 lanes)
typedef __attribute__((ext_vector_type(8))) float v8f;   // 16x16 f32 C/D accumulator
typedef __attribute__((ext_vector_type(4))) float f4;

#define N_FEAT   128
#define HIDDEN   64
#define BATCH    16384
#define WAVES_PER_BLOCK 8   // 256 threads = 8 wave32; one sample per wave

__global__ __launch_bounds__(256, 4)
void hypernet_gemv_wmma(const float* __restrict__ x,        // [B,128]
                        const int*   __restrict__ task_ids, // [B]
                        const float* __restrict__ l1_emb,   // [NT, 128*64] row-major (f, h)
                        const float* __restrict__ l1_bias,  // [NT, 64]
                        const float* __restrict__ l2_emb,   // [NT, 64]
                        const float* __restrict__ l2_bias,  // [NT, 1]
                        float*       __restrict__ out)      // [B, 1]
{
    const int lane = threadIdx.x & 31;
    int b = blockIdx.x * WAVES_PER_BLOCK + (threadIdx.x >> 5);
    b = __builtin_amdgcn_readfirstlane(b);          // uniform per wave -> scalar addressing

    int task = task_ids[b];
    task = __builtin_amdgcn_readfirstlane(task);

    const float* __restrict__ W  = l1_emb + (size_t)task * (N_FEAT * HIDDEN);
    const float* __restrict__ xb = x + (size_t)b * N_FEAT;

    const int  nlo  = lane & 15;                    // N index (h within 16-wide tile)
    const int  fsel = (lane >> 4) << 1;             // 0 for lanes 0-15 (K=0,1), 2 for lanes 16-31 (K=2,3)
    const bool hiHalf = (lane >= 16);

    v8f acc0 = {}, acc1 = {}, acc2 = {}, acc3 = {}; // h0 = 0,16,32,48 chains

    #pragma unroll 4
    for (int ks = 0; ks < N_FEAT / 4; ++ks) {
        const int f0 = ks * 4;

        // A = x[f0..f0+3] broadcast down M. ISA 16x4 f32 A layout:
        //   lanes 0-15: VGPR0=K0, VGPR1=K1 ; lanes 16-31: VGPR0=K2, VGPR1=K3
        const f4 xk = *(const f4*)(xb + f0);        // uniform 16B load
        v2f a;
        a.x = hiHalf ? xk.z : xk.x;
        a.y = hiHalf ? xk.w : xk.y;

        // B = W[f0..f0+3][h0..h0+15]. ISA 4x16 f32 B layout mirrors A:
        //   VGPR0: lanes0-15 row K0, lanes16-31 row K2 ; VGPR1: rows K1/K3.
        // Each load: two fully-used contiguous 64B segments; W streamed exactly once (NT hint).
        const int rb = (f0 + fsel) * HIDDEN + nlo;
        {
            v2f bb;
            bb.x = __builtin_nontemporal_load(&W[rb +  0]);
            bb.y = __builtin_nontemporal_load(&W[rb +  0 + HIDDEN]);
            acc0 = __builtin_amdgcn_wmma_f32_16x16x4_f32(false, a, false, bb, (short)0, acc0, false, false);
        }
        {
            v2f bb;
            bb.x = __builtin_nontemporal_load(&W[rb + 16]);
            bb.y = __builtin_nontemporal_load(&W[rb + 16 + HIDDEN]);
            acc1 = __builtin_amdgcn_wmma_f32_16x16x4_f32(false, a, false, bb, (short)0, acc1, false, false);
        }
        {
            v2f bb;
            bb.x = __builtin_nontemporal_load(&W[rb + 32]);
            bb.y = __builtin_nontemporal_load(&W[rb + 32 + HIDDEN]);
            acc2 = __builtin_amdgcn_wmma_f32_16x16x4_f32(false, a, false, bb, (short)0, acc2, false, false);
        }
        {
            v2f bb;
            bb.x = __builtin_nontemporal_load(&W[rb + 48]);
            bb.y = __builtin_nontemporal_load(&W[rb + 48 + HIDDEN]);
            acc3 = __builtin_amdgcn_wmma_f32_16x16x4_f32(false, a, false, bb, (short)0, acc3, false, false);
        }
    }

    // All M rows of D are identical (A rows were identical), so VGPR0 of each
    // chain holds h[h0 + (lane&15)] on BOTH 16-lane halves.
    const float* __restrict__ b1 = l1_bias + (size_t)task * HIDDEN;
    const float* __restrict__ w2 = l2_emb  + (size_t)task * HIDDEN;

    float partial = 0.0f;
    {
        float h = acc0[0] + b1[ 0 + nlo];
        h = 0.5f * h * (1.0f + erff(h * 0.70710678118654752f));   // exact GELU
        partial += h * w2[ 0 + nlo];
    }
    {
        float h = acc1[0] + b1[16 + nlo];
        h = 0.5f * h * (1.0f + erff(h * 0.70710678118654752f));
        partial += h * w2[16 + nlo];
    }
    {
        float h = acc2[0] + b1[32 + nlo];
        h = 0.5f * h * (1.0f + erff(h * 0.70710678118654752f));
        partial += h * w2[32 + nlo];
    }
    {
        float h = acc3[0] + b1[48 + nlo];
        h = 0.5f * h * (1.0f + erff(h * 0.70710678118654752f));
        partial += h * w2[48 + nlo];
    }

    // Butterfly reduce over the 16-lane group (both halves converge to same sum).
    partial += __shfl_xor(partial, 1);
    partial += __shfl_xor(partial, 2);
    partial += __shfl_xor(partial, 4);
    partial += __shfl_xor(partial, 8);

    if (lane == 0) {
        out[b] = partial + l2_bias[task];
    }
}

extern "C" void kernel_launch(void* const* d_in, const int* in_sizes, int n_in,
                              void* d_out, int out_size, void* d_ws, size_t ws_size,
                              hipStream_t stream) {
    (void)in_sizes; (void)n_in; (void)out_size; (void)d_ws; (void)ws_size;
    const float* x        = (const float*)d_in[0];
    const int*   task_ids = (const int*)  d_in[1];
    const float* l1_emb   = (const float*)d_in[2];
    const float* l1_bias  = (const float*)d_in[3];
    const float* l2_emb   = (const float*)d_in[4];
    const float* l2_bias  = (const float*)d_in[5];
    float*       out      = (float*)d_out;

    dim3 grid(BATCH / WAVES_PER_BLOCK);   // 2048 blocks, no remainder -> no guard, EXEC stays all-ones
    dim3 block(256);
    hipLaunchKernelGGL(hypernet_gemv_wmma, grid, block, 0, stream,
                       x, task_ids, l1_emb, l1_bias, l2_emb, l2_bias, out);
}